// DenseCaps_3384434229639
// MI455X (gfx1250) — compile-verified
//
#include <hip/hip_runtime.h>
#include <math.h>

typedef __attribute__((ext_vector_type(2))) float v2f;
typedef __attribute__((ext_vector_type(8))) float v8f;

#define B_DIM 64
#define O_DIM 10
#define I_DIM 8000
#define D_DIM 16
#define E_DIM 8

#define WAVE_I 40                 // capsule indices per wave
#define WAVES_PER_BLK 4
#define BLK_I (WAVE_I * WAVES_PER_BLK)   // 160 i per block
#define NCHUNK (I_DIM / BLK_I)           // 50
#define SV_ELEMS (B_DIM * O_DIM * D_DIM) // 10240
#define TILE_ELEMS (16 * O_DIM * D_DIM)  // 2560 (one b-group's s tile)

__global__ void zero_kernel(float* __restrict__ p, int n) {
    int idx = blockIdx.x * blockDim.x + threadIdx.x;
    for (; idx < n; idx += gridDim.x * blockDim.x) p[idx] = 0.0f;
}

// 128 threads = 4 waves per block. Block handles one b-group (16 batches) x
// BLK_I capsule indices; each wave takes a WAVE_I sub-range. For each i:
// compute u_hat[o, d, b16] tiles with V_WMMA_F32_16X16X4_F32, derive routing
// coefficients c[b,o,i], accumulate s locally in VGPRs, combine waves via
// ds_add_f32 LDS atomics, then one global-atomic flush per block.
//
// WMMA f32 16x16 D-layout: VGPR r, lanes 0-15 -> (M=r,   N=lane),
//                                   lanes 16-31 -> (M=r+8, N=lane-16).
// We use M = d (within o-tile), N = batch offset within b-group.
__global__ __launch_bounds__(128, 1)
void caps_pass_kernel(const float* __restrict__ x,
                      const float* __restrict__ W,
                      const float* __restrict__ vArg,
                      float* __restrict__ sOut,
                      int mode) {
    __shared__ float sred[TILE_ELEMS];   // [b16][O][D] block accumulator, 10 KB

    const int tid  = threadIdx.x;        // 0..127
    const int lane = tid & 31;
    const int wave = tid >> 5;           // 0..3
    const int l15  = lane & 15;
    const int hl   = lane >> 4;          // 0: d=0..7 half, 1: d=8..15 half
    const int bgroup = blockIdx.x & 3;   // chunk-major ordering for L2 reuse of W
    const int chunk  = blockIdx.x >> 2;
    const int bL = bgroup * 16 + l15;    // this lane's batch
    const int dh = hl * 8;               // this lane's d-half base
    const int i0 = chunk * BLK_I + wave * WAVE_I;

    for (int k = tid; k < TILE_ELEMS; k += 128) sred[k] = 0.0f;

    // Per-lane slice of vArg[b, o, d] matching the WMMA D layout.
    v8f vv[O_DIM];
    if (mode != 0) {
        #pragma unroll
        for (int o = 0; o < O_DIM; ++o) {
            const float* vp = vArg + ((bL * O_DIM) + o) * D_DIM + dh;
            #pragma unroll
            for (int r = 0; r < 8; ++r) vv[o][r] = vp[r];
        }
    }

    v8f sacc[O_DIM];
    #pragma unroll
    for (int o = 0; o < O_DIM; ++o)
        #pragma unroll
        for (int r = 0; r < 8; ++r) sacc[o][r] = 0.0f;

    __syncthreads();

    for (int i = i0; i < i0 + WAVE_I; ++i) {
        // B matrix (4x16 per k-step): B[k, n] = x[b0+n, i, e], e = ks*4 + k.
        // Layout: lanes 0-15 hold K=0,1 (VGPR0,1); lanes 16-31 hold K=2,3.
        const float* xb = x + ((size_t)bL * I_DIM + i) * E_DIM + hl * 2;
        v2f B0 = { xb[0], xb[1] };   // e = hl*2 + {0,1}
        v2f B1 = { xb[4], xb[5] };   // e = 4 + hl*2 + {0,1}

        v8f u[O_DIM];
        #pragma unroll
        for (int o = 0; o < O_DIM; ++o) {
            // A matrix (16x4): A[m=d, k] = W[o, i, d, e]; lanes 0-15: K=0,1; 16-31: K=2,3.
            const float* wb = W + (((size_t)o * I_DIM + i) * D_DIM + l15) * E_DIM + hl * 2;
            v2f A0 = { wb[0], wb[1] };
            v2f A1 = { wb[4], wb[5] };
            v8f acc = {};
            acc = __builtin_amdgcn_wmma_f32_16x16x4_f32(false, A0, false, B0,
                                                        (short)0, acc, false, false);
            acc = __builtin_amdgcn_wmma_f32_16x16x4_f32(false, A1, false, B1,
                                                        (short)0, acc, false, false);
            u[o] = acc;   // u_hat[d = r + dh, b = bL] for this o
        }

        float c[O_DIM];
        if (mode == 0) {
            // First routing iteration: b=0 -> softmax over O is uniform = 1/O.
            #pragma unroll
            for (int o = 0; o < O_DIM; ++o) c[o] = 0.1f;
        } else {
            // logit[o] = sum_d u[o,d] * vArg[b,o,d]; halves joined via lane-16 xor.
            float lg[O_DIM];
            #pragma unroll
            for (int o = 0; o < O_DIM; ++o) {
                float p = 0.0f;
                #pragma unroll
                for (int r = 0; r < 8; ++r) p = fmaf(u[o][r], vv[o][r], p);
                p += __shfl_xor(p, 16, 32);
                lg[o] = p;
            }
            float mx = lg[0];
            #pragma unroll
            for (int o = 1; o < O_DIM; ++o) mx = fmaxf(mx, lg[o]);
            float Z = 0.0f;
            #pragma unroll
            for (int o = 0; o < O_DIM; ++o) { lg[o] = __expf(lg[o] - mx); Z += lg[o]; }
            const float invZ = 1.0f / Z;
            #pragma unroll
            for (int o = 0; o < O_DIM; ++o) c[o] = lg[o] * invZ;
        }

        #pragma unroll
        for (int o = 0; o < O_DIM; ++o)
            #pragma unroll
            for (int r = 0; r < 8; ++r)
                sacc[o][r] = fmaf(c[o], u[o][r], sacc[o][r]);
    }

    // Wave -> block reduction through LDS float atomics (ds_add_f32).
    #pragma unroll
    for (int o = 0; o < O_DIM; ++o) {
        float* sp = &sred[((l15 * O_DIM) + o) * D_DIM + dh];
        #pragma unroll
        for (int r = 0; r < 8; ++r)
            unsafeAtomicAdd(&sp[r], sacc[o][r]);
    }
    __syncthreads();

    // Block -> global flush: s[bgroup tile] += sred (one atomic per element).
    float* gbase = sOut + (size_t)bgroup * TILE_ELEMS;
    for (int k = tid; k < TILE_ELEMS; k += 128)
        unsafeAtomicAdd(&gbase[k], sred[k]);
}

// One thread per (b,o): v = squash(s); optionally vAccum += v (running v0+v1 sum
// used as routing-logit argument), optionally write v to out.
__global__ void squash_kernel(const float* __restrict__ s,
                              float* __restrict__ vAccum,
                              float* __restrict__ out,
                              int doAccum, int doOut) {
    int idx = blockIdx.x * blockDim.x + threadIdx.x;
    if (idx >= B_DIM * O_DIM) return;
    const float* sp = s + idx * D_DIM;
    float sv[D_DIM];
    float n2 = 0.0f;
    #pragma unroll
    for (int d = 0; d < D_DIM; ++d) { sv[d] = sp[d]; n2 = fmaf(sv[d], sv[d], n2); }
    const float n = sqrtf(n2);
    const float scale = n2 / ((1.0f + n2) * (n + 1e-8f));
    #pragma unroll
    for (int d = 0; d < D_DIM; ++d) {
        const float v = scale * sv[d];
        if (doAccum) vAccum[idx * D_DIM + d] += v;
        if (doOut)   out[idx * D_DIM + d] = v;
    }
}

extern "C" void kernel_launch(void* const* d_in, const int* in_sizes, int n_in,
                              void* d_out, int out_size, void* d_ws, size_t ws_size,
                              hipStream_t stream) {
    const float* x = (const float*)d_in[0];   // [64, 8000, 8]
    const float* W = (const float*)d_in[1];   // [10, 8000, 16, 8]
    float* out  = (float*)d_out;              // [64, 10, 16]
    float* sPtr = (float*)d_ws;               // [64,10,16] accumulators
    float* vArg = sPtr + SV_ELEMS;            // running v0 (+v1) sum

    const dim3 passGrid(4 * NCHUNK), passBlock(WAVES_PER_BLK * 32);

    // iter 0: uniform c = 1/O
    zero_kernel<<<40, 256, 0, stream>>>(sPtr, 2 * SV_ELEMS);
    caps_pass_kernel<<<passGrid, passBlock, 0, stream>>>(x, W, vArg, sPtr, 0);
    squash_kernel<<<3, 256, 0, stream>>>(sPtr, vArg, out, 1, 0);   // vArg = v0

    // iter 1: logits = u_hat . v0
    zero_kernel<<<40, 256, 0, stream>>>(sPtr, SV_ELEMS);
    caps_pass_kernel<<<passGrid, passBlock, 0, stream>>>(x, W, vArg, sPtr, 1);
    squash_kernel<<<3, 256, 0, stream>>>(sPtr, vArg, out, 1, 0);   // vArg = v0+v1

    // iter 2: logits = u_hat . (v0+v1); final squash -> output
    zero_kernel<<<40, 256, 0, stream>>>(sPtr, SV_ELEMS);
    caps_pass_kernel<<<passGrid, passBlock, 0, stream>>>(x, W, vArg, sPtr, 2);
    squash_kernel<<<3, 256, 0, stream>>>(sPtr, vArg, out, 0, 1);
}